// DeBut_2dConv_70153995812929
// MI455X (gfx1250) — compile-verified
//
#include <hip/hip_runtime.h>

// ---------------------------------------------------------------------------
// DeBut 3x3 conv, fused:  im2col + composed-butterfly block-GEMM (WMMA f32)
//
// R3*R2*R1 is block-diagonal: 64 blocks of (18 in -> 4 out). Weights are
// composed once on device into 16 groups of [72 rows x 16 cols] (zero-padded
// block diagonal), then one fused kernel runs V_WMMA_F32_16X16X4_F32:
//   per wave: 16 samples (M) x 16 channels (N = one group), K = 72 (18 steps)
// ---------------------------------------------------------------------------

typedef __attribute__((ext_vector_type(2))) float v2f;
typedef __attribute__((ext_vector_type(8))) float v8f;

#define CIN     128
#define HWDIM   56
#define LPOS    3136          // 56*56
#define BATCH   16
#define COUT    256
#define NGROUP  16
#define KROWS   72            // features per group (4 blocks * 18)
#define KSTEPS  18            // 72 / 4  (WMMA K = 4)
#define LDSR    20            // padded LDS row stride (conflict-free)

__device__ __forceinline__ int iclamp(int v, int lo, int hi) {
    return v < lo ? lo : (v > hi ? hi : v);
}

// ---------------------------------------------------------------------------
// Kernel 1: compose the 3 butterfly factors into block weights.
// One thread per output channel o (256). Fully unrolled -> wloc stays in
// registers (all indices compile-time; uses k1==k2, r1==c2, (2c3+d)>>2==c3>>1).
// ---------------------------------------------------------------------------
__global__ void debut_compose_weights(const float* __restrict__ tw,
                                      float* __restrict__ wbuf) {
    int o = blockIdx.x * blockDim.x + threadIdx.x;
    if (o >= COUT) return;
    int k3 = o >> 2;
    int r3 = (o >> 1) & 1;
    int d  = o & 1;

    float wloc[18];
#pragma unroll
    for (int i = 0; i < 18; ++i) wloc[i] = 0.f;

#pragma unroll
    for (int c3 = 0; c3 < 4; ++c3) {
        const int qq = c3 >> 1;               // compile-time: which half-block
        const int k2 = 2 * k3 + qq;
        const int r2 = (c3 * 2 + d) & 3;
        // t3 reshape (64,2,2,4): ((k*2+dg)*2+r)*4+c, transposed [k,r,c,dg]
        float coef3 = tw[6656 + ((k3 * 2 + d) * 2 + r3) * 4 + c3];
#pragma unroll
        for (int c2 = 0; c2 < 4; ++c2) {
            // i1 = k2*4+c2  ->  k1 = k2, r1 = c2
            float coef23 = coef3 * tw[4608 + (k2 * 4 + r2) * 4 + c2];
#pragma unroll
            for (int c1 = 0; c1 < 9; ++c1)
                wloc[qq * 9 + c1] += coef23 * tw[(k2 * 4 + c2) * 9 + c1];
        }
    }

    const int g     = o >> 4;
    const int n     = o & 15;
    const int myblk = k3 & 3;                 // which 18-row band is non-zero
#pragma unroll
    for (int blk = 0; blk < 4; ++blk)
#pragma unroll
        for (int j = 0; j < 18; ++j) {
            float v = (blk == myblk) ? wloc[j] : 0.f;
            wbuf[(g * KROWS + blk * 18 + j) * 16 + n] = v;
        }
}

// ---------------------------------------------------------------------------
// Kernel 2: fused im2col + WMMA block-GEMM + bias + NCHW store.
// 256 threads = 8 waves per block; 392 blocks -> 3136 waves.
// wave -> (group g = id&15, tile-range tgrp = id>>4 owning 16 sample-tiles).
// ---------------------------------------------------------------------------
__global__ __launch_bounds__(256) void debut_conv_wmma(
    const float* __restrict__ x,      // (16,128,56,56)
    const float* __restrict__ wbuf,   // (16 groups, 72, 16)
    const float* __restrict__ bias,   // (256)
    float* __restrict__ out)          // (16,256,56,56)
{
    __shared__ float lds[8][16 * LDSR];

    const int lane = threadIdx.x & 31;
    const int wave = threadIdx.x >> 5;
    const int wid  = blockIdx.x * 8 + wave;
    const int g    = wid & (NGROUP - 1);
    const int tgrp = wid >> 4;            // 0..195
    const int m    = lane & 15;           // sample-in-tile / channel column
    const int hi   = lane >> 4;           // lane half
    const int koff = hi * 2;              // K sub-offset for A/B fragments

    // --- B fragments for all 18 K-steps, resident in registers.
    // f32 4x16 layout: VGPR0 = rows {k0, k0+2} across halves, VGPR1 = {k0+1, k0+3}
    v2f bfrag[KSTEPS];
#pragma unroll
    for (int k = 0; k < KSTEPS; ++k) {
        const float* wp = wbuf + (g * KROWS + k * 4 + koff) * 16 + m;
        bfrag[k].x = wp[0];
        bfrag[k].y = wp[16];
    }
    const float biasv = bias[g * 16 + m];

    // --- group feature window [g*72, g*72+72) spans at most TWO kernel
    //     offsets kk (72-wide window crosses <=1 boundary of 128 channels).
    const int f0g   = g * KROWS;
    const int kk0   = f0g >> 7;                 // first kernel offset
    const int ccb   = f0g & 127;                // first channel within kk0
    const int split = 128 - ccb;                // #features with kk == kk0
    const int kh0 = kk0 / 3,            kw0 = kk0 - kh0 * 3;
    const int kk1 = (kk0 + 1 > 8) ? 8 : kk0 + 1;  // clamped (unused if split>=72)
    const int kh1 = kk1 / 3,            kw1 = kk1 - kh1 * 3;
    // lane-adjusted split threshold: feature t = 4k + j + koff
    const int splk = split - koff;

    for (int i = 0; i < 16; ++i) {
        const int tile = tgrp * 16 + i;
        const int s    = tile * 16 + m;       // sample id for this lane
        const int b    = s / LPOS;            // tiles never cross batch (3136%16==0)
        const int pos  = s - b * LPOS;
        const int oh   = pos / HWDIM;
        const int ow   = pos - oh * HWDIM;
        const float* xb = x + (size_t)b * (CIN * LPOS);

        // two clamped tap base pointers + 0/1 masks (pure dataflow, no branch)
        const int ih0 = oh + kh0 - 1, iw0 = ow + kw0 - 1;
        const int ih1 = oh + kh1 - 1, iw1 = ow + kw1 - 1;
        const float msk0 = ((unsigned)ih0 < (unsigned)HWDIM &&
                            (unsigned)iw0 < (unsigned)HWDIM) ? 1.f : 0.f;
        const float msk1 = ((unsigned)ih1 < (unsigned)HWDIM &&
                            (unsigned)iw1 < (unsigned)HWDIM) ? 1.f : 0.f;
        // fold channel base + lane koff into the pointers so the per-feature
        // offset (4k+j)*LPOS is a compile-time immediate in the load.
        const float* P0 = xb + iclamp(ih0, 0, 55) * HWDIM + iclamp(iw0, 0, 55)
                             + (size_t)(ccb + koff) * LPOS;
        const float* P1 = xb + iclamp(ih1, 0, 55) * HWDIM + iclamp(iw1, 0, 55)
                             + (size_t)(ccb + koff - 128) * LPOS;

        v8f acc = {0.f, 0.f, 0.f, 0.f, 0.f, 0.f, 0.f, 0.f};

#pragma unroll
        for (int k = 0; k < KSTEPS; ++k) {
            v2f a;
#pragma unroll
            for (int j = 0; j < 2; ++j) {
                const int  t4 = k * 4 + j;          // compile-time
                const bool q  = (t4 >= splk);       // loop-invariant per lane
                const float* bp = q ? P1 : P0;
                const float  mk = q ? msk1 : msk0;
                const float  v  = bp[(size_t)t4 * LPOS] * mk;
                if (j == 0) a.x = v; else a.y = v;
            }
            acc = __builtin_amdgcn_wmma_f32_16x16x4_f32(
                false, a, false, bfrag[k], (short)0, acc, false, false);
        }

        // --- transpose through LDS for coalesced NCHW stores ---------------
        // D layout: VGPR v -> (M = v + 8*hi, N = m)
        float* lw = lds[wave];
#pragma unroll
        for (int v = 0; v < 8; ++v)
            lw[(v + hi * 8) * LDSR + m] = acc[v] + biasv;
        __syncthreads();

        // lane reads sample row m, channels {2p + hi}; half-wave = 16
        // consecutive spatial positions of one channel -> 64B coalesced.
        float* outp = out + ((size_t)b * COUT + g * 16) * LPOS + pos;
#pragma unroll
        for (int p = 0; p < 8; ++p) {
            const int ch = p * 2 + hi;
            outp[(size_t)ch * LPOS] = lw[m * LDSR + ch];
        }
        __syncthreads();
    }
}

// ---------------------------------------------------------------------------
extern "C" void kernel_launch(void* const* d_in, const int* in_sizes, int n_in,
                              void* d_out, int out_size, void* d_ws, size_t ws_size,
                              hipStream_t stream) {
    const float* x       = (const float*)d_in[0];   // 16*128*56*56
    const float* twiddle = (const float*)d_in[1];   // 7680
    const float* bias    = (const float*)d_in[2];   // 256
    float*       out     = (float*)d_out;           // 16*256*56*56
    float*       wbuf    = (float*)d_ws;            // 16*72*16 floats = 72KB

    debut_compose_weights<<<1, 256, 0, stream>>>(twiddle, wbuf);

    // 3136 sample-tiles * 16 groups = 3136 waves of work (16 tiles each)
    debut_conv_wmma<<<392, 256, 0, stream>>>(x, wbuf, bias, out);
}